// LinearStateSpaceModel_7808250544204
// MI455X (gfx1250) — compile-verified
//
#include <hip/hip_runtime.h>
#include <math.h>

// ---------------------------------------------------------------------------
// Kalman filter (B=16, T=256, XD=YD=128) for gfx1250 / MI455X.
// Kernel 1: one persistent 256-thread workgroup (8 wave32s), all matrices in
//           LDS (4x 128x132 f32 = 270KB + small state ~ 304KB < 320KB/WGP).
//           128x128 products: each wave owns a 16x128 row-block and drives 8
//           independent v_wmma_f32_16x16x4_f32 accumulator chains (hides the
//           WMMA RAW latency, amortizes A-fragment loads 8x).
// Kernel 2: full-grid broadcast of the unique P_filt slice to all batches
//           (the only real bandwidth in this problem: ~268 MB of stores).
// ---------------------------------------------------------------------------

typedef __attribute__((ext_vector_type(2))) float v2f;
typedef __attribute__((ext_vector_type(8))) float v8f;

#define BB   16
#define TT   256
#define XD   128
#define LDP  132                    // padded LDS row pitch (bank spread)
#define LOG2PI 1.8378770664093453f

#define XOFF  0                         // x_filt: B*T*XD floats
#define POFF  (BB*TT*XD)                // 524288
#define PERB  ((size_t)TT*XD*XD)        // 4194304 floats per batch slice
#define LLOFF (POFF + (size_t)BB*PERB)  // 67633152

// ---- 128x128 = op(A)(128x128) x op(B)(128x128) [+ Cin], row-block/wave ----
// Wave w computes row-block ti=w (16 rows), all 8 column tiles concurrently:
// 8 independent WMMA accumulator chains, shared A fragment per k-step.
__device__ __forceinline__ void mm128(int tid,
                                      const float* A, int lda, bool tA,
                                      const float* Bm, int ldb, bool tB,
                                      const float* Cin, int ldc, float csign,
                                      float* D, int ldd)
{
    const int lane = tid & 31;
    const int ti   = tid >> 5;          // wave id = row block
    const int half = lane >> 4;         // 0: lanes 0-15, 1: lanes 16-31
    const int l16  = lane & 15;
    const int rA   = ti * 16 + l16;     // A row (M) for this lane

    v8f acc[8];
    const v8f z = {};
    #pragma unroll
    for (int tj = 0; tj < 8; ++tj) acc[tj] = z;

    #pragma unroll 2
    for (int kb = 0; kb < 128; kb += 4) {
        const int ka = kb + 2 * half;   // k of fragment component 0
        v2f av;
        if (!tA) { av.x = A[rA * lda + ka]; av.y = A[rA * lda + ka + 1]; }
        else     { av.x = A[ka * lda + rA]; av.y = A[(ka + 1) * lda + rA]; }
        #pragma unroll
        for (int tj = 0; tj < 8; ++tj) {
            const int cB = tj * 16 + l16;
            v2f bv;
            if (!tB) { bv.x = Bm[ka * ldb + cB]; bv.y = Bm[(ka + 1) * ldb + cB]; }
            else     { bv.x = Bm[cB * ldb + ka]; bv.y = Bm[cB * ldb + ka + 1]; }
            acc[tj] = __builtin_amdgcn_wmma_f32_16x16x4_f32(
                          false, av, false, bv, (short)0, acc[tj], false, false);
        }
    }
    // C/D layout: VGPR r -> row = r + 8*half, col = l16 (within tile)
    #pragma unroll
    for (int tj = 0; tj < 8; ++tj) {
        #pragma unroll
        for (int r = 0; r < 8; ++r) {
            const int row = ti * 16 + r + 8 * half;
            const int col = tj * 16 + l16;
            float v = acc[tj][r];
            if (Cin) v = Cin[row * ldc + col] + csign * v;
            D[row * ldd + col] = v;
        }
    }
}

// ---- 16x128 = A(16x128) x op(B)(128x128) [+ Cin], one tile per wave ----
__device__ __forceinline__ void mm16(int tid,
                                     const float* A, int lda,
                                     const float* Bm, int ldb, bool tB,
                                     const float* Cin, int ldc, float csign,
                                     float* D, int ldd)
{
    const int lane = tid & 31;
    const int tj   = tid >> 5;          // wave id = column tile
    const int half = lane >> 4;
    const int l16  = lane & 15;
    const int cB   = tj * 16 + l16;

    v8f acc = {};
    #pragma unroll 4
    for (int kb = 0; kb < 128; kb += 4) {
        const int ka = kb + 2 * half;
        v2f av, bv;
        av.x = A[l16 * lda + ka];  av.y = A[l16 * lda + ka + 1];
        if (!tB) { bv.x = Bm[ka * ldb + cB]; bv.y = Bm[(ka + 1) * ldb + cB]; }
        else     { bv.x = Bm[cB * ldb + ka]; bv.y = Bm[cB * ldb + ka + 1]; }
        acc = __builtin_amdgcn_wmma_f32_16x16x4_f32(
                  false, av, false, bv, (short)0, acc, false, false);
    }
    #pragma unroll
    for (int r = 0; r < 8; ++r) {
        const int row = r + 8 * half;   // batch index b
        const int col = tj * 16 + l16;
        float v = acc[r];
        if (Cin) v = Cin[row * ldc + col] + csign * v;
        D[row * ldd + col] = v;
    }
}

__global__ __launch_bounds__(256)
void kalman_seq(const float* __restrict__ y,   // (B,T,YD)
                const float* __restrict__ Ag,  // (XD,XD)
                const float* __restrict__ Cg,  // (YD,XD)
                const float* __restrict__ Qc,  // (XD,XD)
                const float* __restrict__ Rc,  // (YD,YD)
                const float* __restrict__ x0m, // (XD)
                const float* __restrict__ x0c, // (XD,XD)
                float* __restrict__ out,
                float* __restrict__ ws)
{
    extern __shared__ float sm[];
    float* sP   = sm;                    // 128*132  P_pred / P_f / P_next
    float* sT1  = sP  + 128 * LDP;       // 128*132  T1 = P C^T, later T2
    float* sS   = sT1 + 128 * LDP;       // 128*132  S -> chol L -> Sinv
    float* sK   = sS  + 128 * LDP;       // 128*132  Linv -> K
    float* sx   = sK  + 128 * LDP;       // 16*132   x_pred / x_f
    float* sInn = sx  + 16 * LDP;        // 16*132   innovation
    float* st2  = sInn + 16 * LDP;       // 16*132   innov @ Sinv
    float* sxn  = st2 + 16 * LDP;        // 16*132   x_pred_next
    float* sll  = sxn + 16 * LDP;        // 16       log-likelihood accum
    float* scal = sll + 16;              // 4        [0]=chol diag, [1]=logdet

    const int tid = threadIdx.x;
    float* Qw = ws;                      // Q = Qc Qc^T  (global scratch, L2)
    float* Rw = ws + XD * XD;            // R = Rc Rc^T

    // ---- prologue: Q, R, P0 = chol chol^T; x_pred = x0_mean ----
    mm128(tid, Qc, 128, false, Qc, 128, true, nullptr, 0, 0.f, Qw, 128);
    mm128(tid, Rc, 128, false, Rc, 128, true, nullptr, 0, 0.f, Rw, 128);
    mm128(tid, x0c, 128, false, x0c, 128, true, nullptr, 0, 0.f, sP, LDP);
    for (int i = tid; i < BB * XD; i += 256)
        sx[(i >> 7) * LDP + (i & 127)] = x0m[i & 127];
    if (tid < BB) sll[tid] = 0.f;
    __threadfence();                     // make Qw/Rw visible for later reads
    __syncthreads();

    for (int t = 0; t < TT; ++t) {
        // p1: T1 = P @ C^T
        mm128(tid, sP, LDP, false, Cg, 128, true, nullptr, 0, 0.f, sT1, LDP);
        __syncthreads();
        // p2: S = C @ T1 + R
        mm128(tid, Cg, 128, false, sT1, LDP, false, Rw, 128, 1.f, sS, LDP);
        __syncthreads();
        // stabilize: S = 0.5(S+S^T) + jitter*I
        for (int idx = tid; idx < 128 * 128; idx += 256) {
            int i = idx >> 7, j = idx & 127;
            if (i < j) {
                float v = 0.5f * (sS[i * LDP + j] + sS[j * LDP + i]);
                sS[i * LDP + j] = v;
                sS[j * LDP + i] = v;
            }
        }
        if (tid < 128) sS[tid * LDP + tid] += 1e-6f;
        __syncthreads();
        // p3: in-place Cholesky (lower) of S
        for (int j = 0; j < 128; ++j) {
            if (tid == 0) scal[0] = sqrtf(sS[j * LDP + j]);
            __syncthreads();
            float d = scal[0];
            if (tid == 0) sS[j * LDP + j] = d;
            float invd = 1.0f / d;
            for (int i = j + 1 + tid; i < 128; i += 256)
                sS[i * LDP + j] *= invd;
            __syncthreads();
            int n = 127 - j;
            for (int idx = tid; idx < n * n; idx += 256) {
                int k  = j + 1 + idx / n;
                int i2 = j + 1 + idx % n;
                if (i2 >= k)
                    sS[i2 * LDP + k] -= sS[i2 * LDP + j] * sS[k * LDP + j];
            }
            __syncthreads();
        }
        if (tid == 0) {
            float ld = 0.f;
            for (int i = 0; i < 128; ++i) ld += __logf(sS[i * LDP + i]);
            scal[1] = 2.f * ld;
        }
        // p4: Linv = L^{-1} (128 independent column forward-solves)
        if (tid < 128) {
            const int j = tid;
            for (int i = 0; i < j; ++i) sK[i * LDP + j] = 0.f;
            sK[j * LDP + j] = 1.0f / sS[j * LDP + j];
            for (int i = j + 1; i < 128; ++i) {
                float s = 0.f;
                for (int k = j; k < i; ++k)
                    s += sS[i * LDP + k] * sK[k * LDP + j];
                sK[i * LDP + j] = -s / sS[i * LDP + i];
            }
        }
        __syncthreads();
        // p5: Sinv = Linv^T @ Linv (overwrites S; L dead)
        mm128(tid, sK, LDP, true, sK, LDP, false, nullptr, 0, 0.f, sS, LDP);
        __syncthreads();
        // p6: K = T1 @ Sinv (overwrites Linv)
        mm128(tid, sT1, LDP, false, sS, LDP, false, nullptr, 0, 0.f, sK, LDP);
        __syncthreads();
        // p7: innov = y_t - x_pred @ C^T
        mm16(tid, sx, LDP, Cg, 128, true,
             y + (size_t)t * XD, TT * XD, -1.f, sInn, LDP);
        __syncthreads();
        // p8: t2 = innov @ Sinv
        mm16(tid, sInn, LDP, sS, LDP, false, nullptr, 0, 0.f, st2, LDP);
        __syncthreads();
        // p9: ll_b += -0.5*(logdet + werr_b + 128*log(2pi))
        if (tid < BB) {
            float w = 0.f;
            for (int j = 0; j < 128; ++j)
                w += st2[tid * LDP + j] * sInn[tid * LDP + j];
            sll[tid] += -0.5f * (scal[1] + w + 128.0f * LOG2PI);
        }
        // p10: x_f = x_pred + innov @ K^T (in place over sx)
        mm16(tid, sInn, LDP, sK, LDP, true, sx, LDP, 1.f, sx, LDP);
        __syncthreads();
        // write x_filt[b][t][:]
        for (int idx = tid; idx < BB * XD; idx += 256) {
            int b = idx >> 7, i = idx & 127;
            out[XOFF + (size_t)b * TT * XD + (size_t)t * XD + i] =
                sx[b * LDP + i];
        }
        // p11: P_f = P - K @ T1^T (in place over sP)
        mm128(tid, sK, LDP, false, sT1, LDP, true, sP, LDP, -1.f, sP, LDP);
        __syncthreads();
        // p12: write unique P_filt slice (b=0); kernel 2 broadcasts the rest
        for (int idx = tid; idx < XD * XD; idx += 256) {
            int i = idx >> 7, j = idx & 127;
            out[POFF + (size_t)t * XD * XD + idx] = sP[i * LDP + j];
        }
        // p13: T2 = P_f @ A^T
        mm128(tid, sP, LDP, false, Ag, 128, true, nullptr, 0, 0.f, sT1, LDP);
        __syncthreads();
        // p14: P_next = A @ T2 + Q
        mm128(tid, Ag, 128, false, sT1, LDP, false, Qw, 128, 1.f, sP, LDP);
        // p15: x_pred_next = x_f @ A^T
        mm16(tid, sx, LDP, Ag, 128, true, nullptr, 0, 0.f, sxn, LDP);
        __syncthreads();
        for (int idx = tid; idx < BB * XD; idx += 256) {
            int b = idx >> 7, i = idx & 127;
            sx[b * LDP + i] = sxn[b * LDP + i];
        }
        __syncthreads();
    }
    if (tid < BB) out[LLOFF + tid] = sll[tid];
}

// Broadcast P_filt[0][t] -> P_filt[1..15][t]. 15 * 16MB slices, float4 copies.
__global__ __launch_bounds__(256)
void bcast_P(float* __restrict__ out)
{
    const size_t perV4  = PERB / 4;                    // 1,048,576
    const size_t total  = 15 * perV4;                  // 15,728,640
    size_t idx = (size_t)blockIdx.x * 256 + threadIdx.x;
    if (idx >= total) return;
    size_t b = idx / perV4;                            // 0..14 -> batch 1..15
    size_t r = idx % perV4;
    const float4* src = (const float4*)(out + POFF);
    float4*       dst = (float4*)(out + POFF + (b + 1) * PERB);
    dst[r] = src[r];
}

extern "C" void kernel_launch(void* const* d_in, const int* in_sizes, int n_in,
                              void* d_out, int out_size, void* d_ws, size_t ws_size,
                              hipStream_t stream)
{
    (void)in_sizes; (void)n_in; (void)out_size; (void)ws_size;
    const float* y   = (const float*)d_in[0];
    const float* A   = (const float*)d_in[1];
    const float* C   = (const float*)d_in[2];
    const float* Qc  = (const float*)d_in[3];
    const float* Rc  = (const float*)d_in[4];
    const float* x0m = (const float*)d_in[5];
    const float* x0c = (const float*)d_in[6];
    float* out = (float*)d_out;
    float* ws  = (float*)d_ws;

    // LDS: 4*128*132 + 4*16*132 + 16 + 4 floats = 304,208 bytes (<320KB/WGP)
    const size_t smemBytes =
        (4 * 128 * LDP + 4 * 16 * LDP + 16 + 4) * sizeof(float);

    kalman_seq<<<1, 256, smemBytes, stream>>>(y, A, C, Qc, Rc, x0m, x0c, out, ws);

    const size_t totalV4 = 15 * (PERB / 4);
    const int blocks = (int)((totalV4 + 255) / 256);   // 61,440
    bcast_P<<<blocks, 256, 0, stream>>>(out);
}